// CBAModule_18056042513000
// MI455X (gfx1250) — compile-verified
//
#include <hip/hip_runtime.h>
#include <hip/hip_bf16.h>
#include <math.h>

// ---------------- problem constants ----------------
#define B_    32
#define C_    512
#define CR_   32
#define H_    56
#define W_    56
#define HW_   (H_*W_)        // 3136
#define HW4_  (HW_/4)        // 784
#define CHW_  (C_*HW_)       // 1605632
#define NOUT_ ((long long)B_*CHW_)   // 51380224

typedef __attribute__((ext_vector_type(2))) float v2f;
typedef __attribute__((ext_vector_type(8))) float v8f;

// =====================================================================
// Pass 1: per-(b,c) spatial avg + max pooling.  grid = B*C blocks of 128.
// P layout: rows 0..31 = avg (batch 0..31), rows 32..63 = max.
// =====================================================================
__global__ void cbam_pool_kernel(const float* __restrict__ x, float* __restrict__ P) {
    const int bc  = blockIdx.x;           // b*C + c
    const int tid = threadIdx.x;          // 0..127
    const float4* xp = (const float4*)(x + (size_t)bc * HW_);
    float s = 0.f, m = -INFINITY;
    for (int i = tid; i < HW4_; i += 128) {
        float4 v = xp[i];
        s += v.x + v.y + v.z + v.w;
        m = fmaxf(m, fmaxf(fmaxf(v.x, v.y), fmaxf(v.z, v.w)));
    }
    __shared__ float ss[128];
    __shared__ float sm[128];
    ss[tid] = s; sm[tid] = m;
    __syncthreads();
    for (int off = 64; off > 0; off >>= 1) {
        if (tid < off) {
            ss[tid] += ss[tid + off];
            sm[tid]  = fmaxf(sm[tid], sm[tid + off]);
        }
        __syncthreads();
    }
    if (tid == 0) {
        const int b = bc >> 9;            // /C_
        const int c = bc & (C_ - 1);
        P[b * C_ + c]          = ss[0] * (1.f / (float)HW_);  // avg
        P[(b + B_) * C_ + c]   = sm[0];                        // max
    }
}

// =====================================================================
// Pass 2: shared MLP via fp32 WMMA (V_WMMA_F32_16X16X4_F32).
// One block, 256 threads = 8 waves.
//   Layer1: P[64,512] @ w1^T[512,32] -> relu -> Hs in LDS
//   Layer2: Hs[64,32] @ w2^T[32,512]; ca = sigmoid(avg-path + max-path)
// =====================================================================
__global__ void cbam_mlp_kernel(const float* __restrict__ P,
                                const float* __restrict__ w1, const float* __restrict__ b1,
                                const float* __restrict__ w2, const float* __restrict__ b2,
                                float* __restrict__ ca_out) {
    __shared__ float Hs[64][33];          // padded to dodge bank conflicts
    const int tid   = threadIdx.x;
    const int wave  = tid >> 5;           // 0..7
    const int lane  = tid & 31;
    const int lrow  = lane & 15;
    const int lhalf = lane >> 4;          // 0 or 1

    // ---------- layer 1 ----------
    {
        const int mt = wave >> 1;         // 0..3 (M tile of 64 rows)
        const int nt = wave & 1;          // 0..1 (N tile of 32 cols)
        const int arow = mt * 16 + lrow;  // A: lane holds row M, K pair at 2*lhalf
        const int bcol = nt * 16 + lrow;  // B: lane holds col N, K pair at 2*lhalf
        v8f acc = {0.f,0.f,0.f,0.f,0.f,0.f,0.f,0.f};
        for (int k = 0; k < C_; k += 4) {
            const int kk = k + 2 * lhalf;
            v2f a; a.x = P[arow * C_ + kk];  a.y = P[arow * C_ + kk + 1];
            v2f b; b.x = w1[bcol * C_ + kk]; b.y = w1[bcol * C_ + kk + 1]; // B[k][n]=w1[n][k]
            acc = __builtin_amdgcn_wmma_f32_16x16x4_f32(
                false, a, false, b, (short)0, acc, false, false);
        }
        const int n = nt * 16 + lrow;
        const float bias = b1[n];
        #pragma unroll
        for (int v = 0; v < 8; ++v) {
            const int m = mt * 16 + v + 8 * lhalf;   // C/D layout
            Hs[m][n] = fmaxf(acc[v] + bias, 0.f);
        }
    }
    __syncthreads();

    // ---------- layer 2 (avg + max chains fused before sigmoid) ----------
    for (int t = wave; t < 64; t += 8) {   // uniform across waves: EXEC stays full
        const int mo = t & 1;              // batch tile 0..1
        const int nc = t >> 1;             // channel tile 0..31
        const int n  = nc * 16 + lrow;
        v8f accA = {0.f,0.f,0.f,0.f,0.f,0.f,0.f,0.f};
        v8f accM = {0.f,0.f,0.f,0.f,0.f,0.f,0.f,0.f};
        for (int k = 0; k < CR_; k += 4) {
            const int kk = k + 2 * lhalf;
            v2f aA; aA.x = Hs[mo * 16 + lrow][kk];      aA.y = Hs[mo * 16 + lrow][kk + 1];
            v2f aM; aM.x = Hs[32 + mo * 16 + lrow][kk]; aM.y = Hs[32 + mo * 16 + lrow][kk + 1];
            v2f b;  b.x  = w2[n * CR_ + kk];            b.y  = w2[n * CR_ + kk + 1]; // B[k][n]=w2[n][k]
            accA = __builtin_amdgcn_wmma_f32_16x16x4_f32(
                false, aA, false, b, (short)0, accA, false, false);
            accM = __builtin_amdgcn_wmma_f32_16x16x4_f32(
                false, aM, false, b, (short)0, accM, false, false);
        }
        const float bias2 = 2.f * b2[n];   // +b2 applied inside each mlp() branch
        #pragma unroll
        for (int v = 0; v < 8; ++v) {
            const int m = mo * 16 + v + 8 * lhalf;     // batch index
            const float val = accA[v] + accM[v] + bias2;
            ca_out[m * C_ + n] = 1.f / (1.f + __expf(-val));
        }
    }
}

// =====================================================================
// Pass 3: channel mean/max of xg = x*ca.  One thread per (b,h,w).
// =====================================================================
__global__ void cbam_spatial_stats_kernel(const float* __restrict__ x,
                                          const float* __restrict__ ca,
                                          float* __restrict__ smean,
                                          float* __restrict__ smax) {
    const int idx = blockIdx.x * blockDim.x + threadIdx.x;
    if (idx >= B_ * HW_) return;
    const int b  = idx / HW_;
    const int hw = idx - b * HW_;
    const float* xb  = x + (size_t)b * CHW_ + hw;
    const float* cab = ca + b * C_;
    float s = 0.f, m = -INFINITY;
    for (int c = 0; c < C_; ++c) {
        const float v = xb[(size_t)c * HW_] * cab[c];
        s += v;
        m  = fmaxf(m, v);
    }
    smean[idx] = s * (1.f / (float)C_);
    smax[idx]  = m;
}

// =====================================================================
// Pass 4: 5x5 conv (2-in, 1-out, pad 2) + sigmoid -> sp
// =====================================================================
__global__ void cbam_conv_kernel(const float* __restrict__ smean,
                                 const float* __restrict__ smax,
                                 const float* __restrict__ w3,
                                 const float* __restrict__ b3,
                                 float* __restrict__ sp_out) {
    const int idx = blockIdx.x * blockDim.x + threadIdx.x;
    if (idx >= B_ * HW_) return;
    const int b  = idx / HW_;
    const int hw = idx - b * HW_;
    const int h  = hw / W_;
    const int w  = hw - h * W_;
    float acc = b3[0];
    for (int kh = 0; kh < 5; ++kh) {
        const int hh = h + kh - 2;
        if (hh < 0 || hh >= H_) continue;
        for (int kw = 0; kw < 5; ++kw) {
            const int ww = w + kw - 2;
            if (ww < 0 || ww >= W_) continue;
            const int off = b * HW_ + hh * W_ + ww;
            acc += smean[off] * w3[kh * 5 + kw] + smax[off] * w3[25 + kh * 5 + kw];
        }
    }
    sp_out[idx] = 1.f / (1.f + __expf(-acc));
}

// =====================================================================
// Pass 5: out = x * ca[b,c] * sp[b,hw], float4-vectorized.
// =====================================================================
__global__ void cbam_final_kernel(const float* __restrict__ x,
                                  const float* __restrict__ ca,
                                  const float* __restrict__ sp,
                                  float* __restrict__ out) {
    const int i = blockIdx.x * blockDim.x + threadIdx.x;   // float4 index
    if (i >= (int)(NOUT_ / 4)) return;
    const int per_b = C_ * HW4_;          // 401408 float4 per batch
    const int b   = i / per_b;
    const int rem = i - b * per_b;
    const int c   = rem / HW4_;
    const int hw4 = rem - c * HW4_;
    const float4 xv  = ((const float4*)x)[i];
    const float  cav = ca[b * C_ + c];
    const float4 spv = ((const float4*)sp)[b * HW4_ + hw4];
    float4 o;
    o.x = xv.x * cav * spv.x;
    o.y = xv.y * cav * spv.y;
    o.z = xv.z * cav * spv.z;
    o.w = xv.w * cav * spv.w;
    ((float4*)out)[i] = o;
}

// =====================================================================
extern "C" void kernel_launch(void* const* d_in, const int* in_sizes, int n_in,
                              void* d_out, int out_size, void* d_ws, size_t ws_size,
                              hipStream_t stream) {
    const float* x  = (const float*)d_in[0];
    const float* w1 = (const float*)d_in[1];
    const float* b1 = (const float*)d_in[2];
    const float* w2 = (const float*)d_in[3];
    const float* b2 = (const float*)d_in[4];
    const float* w3 = (const float*)d_in[5];
    const float* b3 = (const float*)d_in[6];

    float* out_f = (float*)d_out;
    float* ca    = out_f + NOUT_;             // [B,C]    in output slot 2
    float* sp    = ca + (B_ * C_);            // [B,1,H,W] in output slot 3

    // workspace: P[64,512] | smean[B,HW] | smax[B,HW]
    float* P     = (float*)d_ws;
    float* smean = P + 64 * C_;
    float* smax  = smean + B_ * HW_;

    // P1: pooling
    cbam_pool_kernel<<<B_ * C_, 128, 0, stream>>>(x, P);
    // P2: WMMA MLP -> ca (written straight into d_out's ca slice)
    cbam_mlp_kernel<<<1, 256, 0, stream>>>(P, w1, b1, w2, b2, ca);
    // P3: channel mean/max of xg
    cbam_spatial_stats_kernel<<<(B_ * HW_ + 255) / 256, 256, 0, stream>>>(x, ca, smean, smax);
    // P4: 5x5 conv + sigmoid -> sp (into d_out's sp slice)
    cbam_conv_kernel<<<(B_ * HW_ + 255) / 256, 256, 0, stream>>>(smean, smax, w3, b3, sp);
    // P5: out = x * ca * sp
    const int n4 = (int)(NOUT_ / 4);
    cbam_final_kernel<<<(n4 + 255) / 256, 256, 0, stream>>>(x, ca, sp, out_f);
}